// SAMixedScore_MultiHeadAttention_70660801954158
// MI455X (gfx1250) — compile-verified
//
#include <hip/hip_runtime.h>
#include <hip/hip_bf16.h>

// B=8, H=8, R=128, C=128, DQ=16, MS=16
#define Bb 8
#define Hh 8
#define Rr 128
#define Cc 128
#define DQ 16
#define MS 16

typedef __attribute__((ext_vector_type(2))) float v2f;
typedef __attribute__((ext_vector_type(8))) float v8f;

__device__ __forceinline__ v8f wmma_f32(v2f a, v2f b, v8f c) {
    // V_WMMA_F32_16X16X4_F32: D = A(16x4 f32) * B(4x16 f32) + C(16x16 f32)
    return __builtin_amdgcn_wmma_f32_16x16x4_f32(
        /*neg_a=*/false, a, /*neg_b=*/false, b,
        /*c_mod=*/(short)0, c, /*reuse_a=*/false, /*reuse_b=*/false);
}

__device__ __forceinline__ float gelu_exact(float x) {
    return 0.5f * x * (1.0f + erff(x * 0.70710678118654752f));
}

// ---------------------------------------------------------------------------
// Stage 1: per (b,h)  -> score[b,h,r,c] = (softmax(gelu(qk_s^T/4)) @ relu(qk_a^T/4)) / 4
// grid = B*H blocks, 256 threads (8 waves), dynamic LDS = 2*128*128*4 = 128KB
// ---------------------------------------------------------------------------
__global__ void __launch_bounds__(256) stage1_kernel(
    const float* __restrict__ q, const float* __restrict__ k_s,
    const float* __restrict__ k_a, float* __restrict__ score)
{
    const int bh   = blockIdx.x;
    const int tid  = threadIdx.x;
    const int wave = tid >> 5;
    const int lane = tid & 31;
    const int ll   = lane & 15;
    const int lh   = lane >> 4;
    const int mBase = wave * 16;

    extern __shared__ float smem[];
    float* wself = smem;                 // [128][128]
    float* dpa   = smem + Rr * Rr;       // [128][128]

    const float* qb  = q   + bh * Rr * DQ;
    const float* ksb = k_s + bh * Rr * DQ;
    const float* kab = k_a + bh * Cc * DQ;

    // ---- dps = gelu((q @ k_s^T) / 4) -> wself LDS
    for (int nb = 0; nb < 8; ++nb) {
        const int nBase = nb * 16;
        v8f acc = {};
        for (int kk = 0; kk < 4; ++kk) {
            const int k0 = kk * 4 + 2 * lh;
            v2f a, bm;
            a.x  = qb[(mBase + ll) * DQ + k0];
            a.y  = qb[(mBase + ll) * DQ + k0 + 1];
            bm.x = ksb[(nBase + ll) * DQ + k0];
            bm.y = ksb[(nBase + ll) * DQ + k0 + 1];
            acc = wmma_f32(a, bm, acc);
        }
        for (int k = 0; k < 8; ++k)
            wself[(mBase + k + 8 * lh) * Rr + nBase + ll] = gelu_exact(acc[k] * 0.25f);
    }
    __syncthreads();

    // ---- row softmax of wself (threads 0..127, one row each)
    if (tid < Rr) {
        float* row = wself + tid * Rr;
        float mx = row[0];
        for (int i = 1; i < Rr; ++i) mx = fmaxf(mx, row[i]);
        float s = 0.f;
        for (int i = 0; i < Rr; ++i) { float e = __expf(row[i] - mx); row[i] = e; s += e; }
        float inv = 1.0f / s;
        for (int i = 0; i < Rr; ++i) row[i] *= inv;
    }
    __syncthreads();

    // ---- dpa = relu((q @ k_a^T) / 4) -> dpa LDS
    for (int nb = 0; nb < 8; ++nb) {
        const int nBase = nb * 16;
        v8f acc = {};
        for (int kk = 0; kk < 4; ++kk) {
            const int k0 = kk * 4 + 2 * lh;
            v2f a, bm;
            a.x  = qb[(mBase + ll) * DQ + k0];
            a.y  = qb[(mBase + ll) * DQ + k0 + 1];
            bm.x = kab[(nBase + ll) * DQ + k0];
            bm.y = kab[(nBase + ll) * DQ + k0 + 1];
            acc = wmma_f32(a, bm, acc);
        }
        for (int k = 0; k < 8; ++k)
            dpa[(mBase + k + 8 * lh) * Cc + nBase + ll] = fmaxf(acc[k] * 0.25f, 0.f);
    }
    __syncthreads();

    // ---- score = (wself @ dpa) / 4 -> global
    float* outp = score + bh * Rr * Cc;
    for (int nb = 0; nb < 8; ++nb) {
        const int nBase = nb * 16;
        v8f acc = {};
        for (int kk = 0; kk < 32; ++kk) {
            const int k0 = kk * 4 + 2 * lh;
            v2f a, bm;
            a.x  = wself[(mBase + ll) * Rr + k0];
            a.y  = wself[(mBase + ll) * Rr + k0 + 1];
            bm.x = dpa[k0 * Cc + nBase + ll];
            bm.y = dpa[(k0 + 1) * Cc + nBase + ll];
            acc = wmma_f32(a, bm, acc);
        }
        for (int k = 0; k < 8; ++k)
            outp[(mBase + k + 8 * lh) * Cc + nBase + ll] = acc[k] * 0.25f;
    }
}

// ---------------------------------------------------------------------------
// Stage 2: per (b,r,h) tile. grid = B*R*H = 8192 blocks, 256 threads.
// dynamic LDS = (2048 + 2048 + 16384 + 16 + 16 + 128 + 2) floats ~= 82.6KB
// ---------------------------------------------------------------------------
__global__ void __launch_bounds__(256) stage2_kernel(
    const float* __restrict__ score, const float* __restrict__ cost,
    const float* __restrict__ v,
    const float* __restrict__ mix1_w, const float* __restrict__ mix1_b,
    const float* __restrict__ Waa,
    const float* __restrict__ mix2_w, const float* __restrict__ mix2_b,
    const float* __restrict__ edge_w,
    const float* __restrict__ ln_g, const float* __restrict__ ln_b,
    float* __restrict__ out_attn, float* __restrict__ edge_part)
{
    const int idx = blockIdx.x;
    const int h = idx & (Hh - 1);
    const int r = (idx >> 3) & (Rr - 1);
    const int b = idx >> 10;

    const int tid  = threadIdx.x;
    const int wave = tid >> 5;
    const int lane = tid & 31;
    const int ll   = lane & 15;
    const int lh   = lane >> 4;
    const int mBase = wave * 16;

    extern __shared__ float smem[];
    float* ms1  = smem;                          // [128][16]
    float* tbuf = smem + Cc * MS;                // [128][16]  (t, then ae)
    float* dd   = smem + 2 * Cc * MS;            // [128][128]
    float* mu_s = dd + Cc * Cc;                  // [16]
    float* rs_s = mu_s + MS;                     // [16]
    float* msc  = rs_s + MS;                     // [128]
    float* red  = msc + Cc;                      // [2]

    const float* srow = score + ((b * Hh + h) * Rr + r) * Cc;
    const float* crow = cost + (b * Rr + r) * Cc;
    const float* m1w0 = mix1_w + h * 2 * MS;
    const float* m1w1 = m1w0 + MS;
    const float* m1b  = mix1_b + h * MS;

    // ---- ms1[c][e] = relu(score*w0 + cost*w1 + b)
    for (int i = tid; i < Cc * MS; i += 256) {
        const int c = i >> 4, e = i & 15;
        ms1[i] = fmaxf(srow[c] * m1w0[e] + crow[c] * m1w1[e] + m1b[e], 0.f);
    }
    __syncthreads();

    // ---- t = ms1 @ Waa[h]   (128x16 @ 16x16), one 16x16 tile per wave
    {
        const float* W = Waa + h * MS * MS;
        v8f acc = {};
        for (int kk = 0; kk < 4; ++kk) {
            const int k0 = kk * 4 + 2 * lh;
            v2f a, bm;
            a.x  = ms1[(mBase + ll) * MS + k0];
            a.y  = ms1[(mBase + ll) * MS + k0 + 1];
            bm.x = W[k0 * MS + ll];
            bm.y = W[(k0 + 1) * MS + ll];
            acc = wmma_f32(a, bm, acc);
        }
        for (int k = 0; k < 8; ++k)
            tbuf[(mBase + k + 8 * lh) * MS + ll] = acc[k];
    }
    __syncthreads();

    // ---- dd = t @ ms1^T  (128x16 @ 16x128) -> LDS
    for (int nb = 0; nb < 8; ++nb) {
        const int nBase = nb * 16;
        v8f acc = {};
        for (int kk = 0; kk < 4; ++kk) {
            const int k0 = kk * 4 + 2 * lh;
            v2f a, bm;
            a.x  = tbuf[(mBase + ll) * MS + k0];
            a.y  = tbuf[(mBase + ll) * MS + k0 + 1];
            bm.x = ms1[(nBase + ll) * MS + k0];
            bm.y = ms1[(nBase + ll) * MS + k0 + 1];
            acc = wmma_f32(a, bm, acc);
        }
        for (int k = 0; k < 8; ++k)
            dd[(mBase + k + 8 * lh) * Cc + nBase + ll] = acc[k];
    }
    __syncthreads();

    // ---- dw = softmax(dd, axis=-1) in place
    if (tid < Cc) {
        float* row = dd + tid * Cc;
        float mx = row[0];
        for (int i = 1; i < Cc; ++i) mx = fmaxf(mx, row[i]);
        float s = 0.f;
        for (int i = 0; i < Cc; ++i) { float e = __expf(row[i] - mx); row[i] = e; s += e; }
        float inv = 1.0f / s;
        for (int i = 0; i < Cc; ++i) row[i] *= inv;
    }
    __syncthreads();

    // ---- ae = dw @ ms1 + ms1 -> tbuf (reuse)
    {
        v8f acc = {};
        for (int kk = 0; kk < 32; ++kk) {
            const int k0 = kk * 4 + 2 * lh;
            v2f a, bm;
            a.x  = dd[(mBase + ll) * Cc + k0];
            a.y  = dd[(mBase + ll) * Cc + k0 + 1];
            bm.x = ms1[k0 * MS + ll];
            bm.y = ms1[(k0 + 1) * MS + ll];
            acc = wmma_f32(a, bm, acc);
        }
        for (int k = 0; k < 8; ++k) {
            const int row = mBase + k + 8 * lh;
            tbuf[row * MS + ll] = acc[k] + ms1[row * MS + ll];
        }
    }
    __syncthreads();

    // ---- LayerNorm over the c axis (length 128), per channel e
    if (tid < MS) {
        float s = 0.f, s2 = 0.f;
        for (int c = 0; c < Cc; ++c) { float x = tbuf[c * MS + tid]; s += x; s2 += x * x; }
        const float mu = s * (1.0f / Cc);
        const float var = s2 * (1.0f / Cc) - mu * mu;
        mu_s[tid] = mu;
        rs_s[tid] = rsqrtf(var + 1e-5f);
    }
    __syncthreads();
    for (int i = tid; i < Cc * MS; i += 256) {
        const int c = i >> 4, e = i & 15;
        tbuf[i] = (tbuf[i] - mu_s[e]) * rs_s[e] * ln_g[c] + ln_b[c];
    }
    __syncthreads();

    // ---- edge partial for this head -> workspace (deterministic, no atomics)
    if (tid < Cc) {
        const float* ew = edge_w + h * MS;
        float p = 0.f;
        for (int e = 0; e < MS; ++e) p += tbuf[tid * MS + e] * ew[e];
        edge_part[((b * Rr + r) * Hh + h) * Cc + tid] = p;
    }

    // ---- mixed score ms2 -> softmax over c -> out = w @ v
    if (tid < Cc) {
        const float* mw = mix2_w + h * MS;
        float s = mix2_b[h];
        for (int e = 0; e < MS; ++e) s += ms1[tid * MS + e] * mw[e];
        msc[tid] = s;
    }
    __syncthreads();
    if (tid == 0) {
        float mx = msc[0];
        for (int i = 1; i < Cc; ++i) mx = fmaxf(mx, msc[i]);
        float s = 0.f;
        for (int i = 0; i < Cc; ++i) s += __expf(msc[i] - mx);
        red[0] = mx;
        red[1] = 1.0f / s;
    }
    __syncthreads();
    if (tid < Cc) msc[tid] = __expf(msc[tid] - red[0]) * red[1];
    __syncthreads();
    if (tid < DQ) {
        const float* vb = v + (b * Hh + h) * Cc * DQ;
        float o = 0.f;
        for (int c = 0; c < Cc; ++c) o += msc[c] * vb[c * DQ + tid];
        out_attn[(b * Rr + r) * (Hh * DQ) + h * DQ + tid] = o;
    }
}

// ---------------------------------------------------------------------------
// Stage 3: edge2[b,r,c] = sum_h edge_part[b,r,h,c] + edge_b
// ---------------------------------------------------------------------------
__global__ void __launch_bounds__(256) edge_reduce_kernel(
    const float* __restrict__ edge_part, const float* __restrict__ edge_b,
    float* __restrict__ out_edge)
{
    const int i = blockIdx.x * blockDim.x + threadIdx.x;  // over B*R*C
    if (i >= Bb * Rr * Cc) return;
    const int c  = i & (Cc - 1);
    const int br = i >> 7;
    float s = edge_b[0];
    for (int h = 0; h < Hh; ++h) s += edge_part[(br * Hh + h) * Cc + c];
    out_edge[i] = s;
}

extern "C" void kernel_launch(void* const* d_in, const int* in_sizes, int n_in,
                              void* d_out, int out_size, void* d_ws, size_t ws_size,
                              hipStream_t stream) {
    const float* q       = (const float*)d_in[0];
    const float* k_s     = (const float*)d_in[1];
    const float* k_a     = (const float*)d_in[2];
    const float* v       = (const float*)d_in[3];
    const float* cost    = (const float*)d_in[4];
    const float* mix1_w  = (const float*)d_in[5];
    const float* mix1_b  = (const float*)d_in[6];
    const float* Waa     = (const float*)d_in[7];
    const float* mix2_w  = (const float*)d_in[8];
    const float* mix2_b  = (const float*)d_in[9];
    const float* edge_w  = (const float*)d_in[10];
    const float* edge_b  = (const float*)d_in[11];
    const float* ln_g    = (const float*)d_in[12];
    const float* ln_b    = (const float*)d_in[13];

    float* out_attn = (float*)d_out;                       // [B,R,H*DQ]
    float* out_edge = out_attn + Bb * Rr * Hh * DQ;        // [B,R,C]

    float* score     = (float*)d_ws;                       // [B,H,R,C]   4MB
    float* edge_part = score + Bb * Hh * Rr * Cc;          // [B,R,H,C]   4MB

    // Stage 1: 64 blocks, 128KB dynamic LDS (wself + dpa)
    const size_t lds1 = (size_t)(2 * Rr * Cc) * sizeof(float);
    stage1_kernel<<<Bb * Hh, 256, lds1, stream>>>(q, k_s, k_a, score);

    // Stage 2: 8192 blocks, ~82.6KB dynamic LDS
    const size_t lds2 = (size_t)(2 * Cc * MS + Cc * Cc + MS + MS + Cc + 2) * sizeof(float);
    stage2_kernel<<<Bb * Rr * Hh, 256, lds2, stream>>>(
        score, cost, v, mix1_w, mix1_b, Waa, mix2_w, mix2_b,
        edge_w, ln_g, ln_b, out_attn, edge_part);

    // Stage 3: edge reduction over heads
    edge_reduce_kernel<<<(Bb * Rr * Cc + 255) / 256, 256, 0, stream>>>(
        edge_part, edge_b, out_edge);
}